// LinearRecurrentUnit_41214506172659
// MI455X (gfx1250) — compile-verified
//
#include <hip/hip_runtime.h>

#define HDIM 64
#define XDIM 128
#define TLEN 8192
#define LCHUNK 64
#define NCHUNK 128   // TLEN / LCHUNK
#define PSTRIDE (HDIM + 4)   // padded LDS stride: bank = (4*row + col) % 64, conflict-free

typedef float v2f __attribute__((ext_vector_type(2)));
typedef float v8f __attribute__((ext_vector_type(8)));

// D = A(16x4) * B(4x16) + C  using V_WMMA_F32_16X16X4_F32 (fp32, full precision)
__device__ __forceinline__ v8f wmma4(v2f a, v2f b, v8f c) {
    return __builtin_amdgcn_wmma_f32_16x16x4_f32(
        /*neg_a=*/false, a, /*neg_b=*/false, b,
        /*c_mod=*/(short)0, c, /*reuse_a=*/false, /*reuse_b=*/false);
}

// ---- CDNA5 async global->LDS copy (ASYNCcnt path) ----
// generic LDS pointer: low 32 bits are the LDS byte offset (aperture in [63:32])
__device__ __forceinline__ void async_ld_b128(void* lds_ptr, const float* gsrc) {
    uint32_t ldsoff = (uint32_t)(uintptr_t)lds_ptr;
    asm volatile("global_load_async_to_lds_b128 %0, %1, off"
                 :: "v"(ldsoff), "v"(gsrc) : "memory");
}
__device__ __forceinline__ void wait_async0() {
    asm volatile("s_wait_asynccnt 0x0" ::: "memory");
}
// stage a 64x64 f32 matrix (row-major) into padded LDS [64][PSTRIDE]; 256 threads,
// 4x b128 chunks per thread (each 16B chunk stays within one row: 16 chunks/row)
__device__ __forceinline__ void stage64(const float* __restrict__ src,
                                        float (*dst)[PSTRIDE], int tid) {
    for (int t = 0; t < 4; ++t) {
        int chunk = tid + t * 256;        // 1024 chunks of 4 floats
        int n = chunk >> 4;               // row
        int k = (chunk & 15) << 2;        // col
        async_ld_b128(&dst[n][k], src + (chunk << 2));
    }
}

// ---- WMMA fragment helpers (ISA 7.12.2 layouts, wave32) ----
__device__ __forceinline__ v2f ldA(const float* src, int ld, int m0, int k0,
                                   int r16, int hi) {
    const float* p = src + (size_t)(m0 + r16) * ld + k0 + 2 * hi;
    v2f a; a.x = p[0]; a.y = p[1]; return a;
}
__device__ __forceinline__ v2f ldB(const float* src, int ld, int k0, int n0,
                                   int r16, int hi) {
    const float* p = src + (size_t)(k0 + 2 * hi) * ld + n0 + r16;
    v2f b; b.x = p[0]; b.y = p[ld]; return b;
}
// B fragment of M^T where M is row-major [N x K]: opB[k][n] = M[n][k]
__device__ __forceinline__ v2f ldBT(const float* src, int ld, int k0, int n0,
                                    int r16, int hi) {
    const float* p = src + (size_t)(n0 + r16) * ld + k0 + 2 * hi;
    v2f b; b.x = p[0]; b.y = p[1]; return b;
}
__device__ __forceinline__ void stD(float* dst, int ld, int m0, int n0,
                                    int r16, int hi, v8f acc) {
    for (int j = 0; j < 8; ++j)
        dst[(size_t)(m0 + j + 8 * hi) * ld + n0 + r16] = acc[j];
}
__device__ __forceinline__ v8f ldC(const float* src, int ld, int m0, int n0,
                                   int r16, int hi) {
    v8f c;
    for (int j = 0; j < 8; ++j)
        c[j] = src[(size_t)(m0 + j + 8 * hi) * ld + n0 + r16];
    return c;
}

// ============================================================================
// K0: build A, powers P[0..64] (P[k]=A^k) by doubling, Q[k]=A^(64*2^k) k=0..6
// ============================================================================
__global__ __launch_bounds__(256) void lru_k0_powers(
    const float* __restrict__ A_raw, const float* __restrict__ h0,
    float* __restrict__ P, float* __restrict__ Q, float* __restrict__ g) {
    const int tid = threadIdx.x;
    const int lane = tid & 31, w = tid >> 5;
    const int r16 = lane & 15, hi = lane >> 4;

    for (int i = tid; i < HDIM * HDIM; i += blockDim.x) {
        int r = i >> 6, c = i & 63;
        float id = (r == c) ? 1.0f : 0.0f;
        P[i] = id;
        P[HDIM * HDIM + i] = 0.9f * id + 0.1f * A_raw[i];
    }
    if (tid < HDIM) g[tid] = h0[tid];
    __threadfence(); __syncthreads();

    // doubling: for m in {1,2,4,8,16,32}: P[m+j] = P[m] @ P[j], j = 1..m
    for (int m = 1; m <= 32; m <<= 1) {
        const int jobs = m * 16;
        for (int job = w; job < jobs; job += 8) {
            const int j = 1 + (job >> 4);
            const int tile = job & 15;
            const int m0 = (tile >> 2) * 16, n0 = (tile & 3) * 16;
            const float* Am = P + (size_t)m * HDIM * HDIM;
            const float* Bj = P + (size_t)j * HDIM * HDIM;
            float* D = P + (size_t)(m + j) * HDIM * HDIM;
            v8f acc = {};
            for (int kk = 0; kk < 16; ++kk) {
                v2f a = ldA(Am, HDIM, m0, kk * 4, r16, hi);
                v2f b = ldB(Bj, HDIM, kk * 4, n0, r16, hi);
                acc = wmma4(a, b, acc);
            }
            stD(D, HDIM, m0, n0, r16, hi, acc);
        }
        __threadfence(); __syncthreads();
    }

    // Q[0] = P[64]; Q[k] = Q[k-1]^2
    for (int i = tid; i < HDIM * HDIM; i += blockDim.x)
        Q[i] = P[(size_t)64 * HDIM * HDIM + i];
    __threadfence(); __syncthreads();
    for (int k = 1; k < 7; ++k) {
        const float* S = Q + (size_t)(k - 1) * HDIM * HDIM;
        float* D = Q + (size_t)k * HDIM * HDIM;
        for (int tile = w; tile < 16; tile += 8) {
            const int m0 = (tile >> 2) * 16, n0 = (tile & 3) * 16;
            v8f acc = {};
            for (int kk = 0; kk < 16; ++kk) {
                v2f a = ldA(S, HDIM, m0, kk * 4, r16, hi);
                v2f b = ldB(S, HDIM, kk * 4, n0, r16, hi);
                acc = wmma4(a, b, acc);
            }
            stD(D, HDIM, m0, n0, r16, hi, acc);
        }
        __threadfence(); __syncthreads();
    }
}

// ============================================================================
// K1: per-chunk  b = x_chunk @ B^T + c  (WMMA, K=128), then local scan as
// triangular convolution  Y[r,:] = sum_{d<=r} b[r-d,:] @ (A^d)^T  (WMMA, K=64)
// P[d] double-buffered in LDS via async global->LDS copies (ASYNCcnt).
// ============================================================================
__global__ __launch_bounds__(256) void lru_k1_local(
    const float* __restrict__ x, const float* __restrict__ Bm,
    const float* __restrict__ cvec, const float* __restrict__ P,
    float* __restrict__ Y, float* __restrict__ g) {
    const int c = blockIdx.x;
    const int tid = threadIdx.x;
    const int lane = tid & 31, w = tid >> 5;
    const int r16 = lane & 15, hi = lane >> 4;

    __shared__ float bLDS[LCHUNK][PSTRIDE];     // 17.4 KB
    __shared__ float pLDS[2][HDIM][PSTRIDE];    // 34.8 KB double buffer for P[d]

    const int m0 = (w & 3) * 16;     // timestep stripe
    const int n0 = (w >> 2) * 32;    // feature half
    const float* xc = x + (size_t)c * LCHUNK * XDIM;

    // kick off async stage of P[0] while doing the b-GEMM
    stage64(P, pLDS[0], tid);

    // phase 1: b = x_chunk[64,128] @ B^T[128,64] + c   -> LDS
    {
        v8f acc0 = {}, acc1 = {};
        for (int kk = 0; kk < XDIM / 4; ++kk) {
            v2f a  = ldA(xc, XDIM, m0, kk * 4, r16, hi);
            v2f b0 = ldBT(Bm, XDIM, kk * 4, n0, r16, hi);
            v2f b1 = ldBT(Bm, XDIM, kk * 4, n0 + 16, r16, hi);
            acc0 = wmma4(a, b0, acc0);
            acc1 = wmma4(a, b1, acc1);
        }
        const float c0 = cvec[n0 + r16], c1 = cvec[n0 + 16 + r16];
        for (int j = 0; j < 8; ++j) { acc0[j] += c0; acc1[j] += c1; }
        stD(&bLDS[0][0], PSTRIDE, m0, n0, r16, hi, acc0);
        stD(&bLDS[0][0], PSTRIDE, m0, n0 + 16, r16, hi, acc1);
    }
    wait_async0();
    __syncthreads();   // bLDS and pLDS[0] ready

    // phase 2: WG-uniform d loop, double-buffered P[d]
    v8f acc0 = {}, acc1 = {};
    const int dmax = m0 + 15;
    for (int d = 0; d < LCHUNK; ++d) {
        const int cur = d & 1;
        if (d + 1 < LCHUNK)   // prefetch next power into the other buffer
            stage64(P + (size_t)(d + 1) * HDIM * HDIM, pLDS[1 - cur], tid);
        if (d <= dmax) {      // wave-uniform: EXEC stays all-ones around WMMA
            const float* Pd = &pLDS[cur][0][0];
            for (int kk = 0; kk < 16; ++kk) {
                const int row = m0 + r16 - d;
                v2f a;
                a.x = (row >= 0) ? bLDS[row][kk * 4 + 2 * hi]     : 0.0f;
                a.y = (row >= 0) ? bLDS[row][kk * 4 + 2 * hi + 1] : 0.0f;
                v2f b0 = ldBT(Pd, PSTRIDE, kk * 4, n0, r16, hi);
                v2f b1 = ldBT(Pd, PSTRIDE, kk * 4, n0 + 16, r16, hi);
                acc0 = wmma4(a, b0, acc0);
                acc1 = wmma4(a, b1, acc1);
            }
        }
        wait_async0();
        __syncthreads();      // next buffer visible to all waves
    }

    float* Yc = Y + (size_t)c * LCHUNK * HDIM;
    stD(Yc, HDIM, m0, n0, r16, hi, acc0);
    stD(Yc, HDIM, m0, n0 + 16, r16, hi, acc1);

    // chunk-end state e_c = Y[63,:] -> g[c+1] (row 63 = stripe 48, j=7, hi=1)
    if (c < NCHUNK - 1 && m0 == 48 && hi == 1) {
        g[(size_t)(c + 1) * HDIM + n0 + r16]      = acc0[7];
        g[(size_t)(c + 1) * HDIM + n0 + 16 + r16] = acc1[7];
    }
}

// ============================================================================
// K2: log-depth carry scan over g[128,64]:
//   7 rounds of  C[c,:] += C[c-2^k,:] @ Q[k]^T   (rows c < 2^k unchanged)
// Q[k] staged to LDS per round via async copies.
// ============================================================================
__global__ __launch_bounds__(256) void lru_k2_carry(
    const float* __restrict__ Q, const float* __restrict__ g,
    float* __restrict__ carr) {
    __shared__ float Cb[NCHUNK][PSTRIDE];      // 34.8 KB, padded (A-frag reads)
    __shared__ float qLDS[HDIM][PSTRIDE];      // 17.4 KB
    const int tid = threadIdx.x;
    const int lane = tid & 31, w = tid >> 5;
    const int r16 = lane & 15, hi = lane >> 4;

    for (int i = tid; i < NCHUNK * HDIM; i += blockDim.x)
        Cb[i >> 6][i & 63] = g[i];
    __syncthreads();

    const int m0 = w * 16;
    for (int k = 0; k < 7; ++k) {
        const int off = 1 << k;
        stage64(Q + (size_t)k * HDIM * HDIM, qLDS, tid);   // async issue
        v8f acc[4];
        for (int nt = 0; nt < 4; ++nt)      // identity part (pre-barrier reads OK)
            acc[nt] = ldC(&Cb[0][0], PSTRIDE, m0, nt * 16, r16, hi);
        wait_async0();
        __syncthreads();                    // qLDS ready
        for (int kk = 0; kk < 16; ++kk) {
            const int row = m0 + r16 - off;
            v2f a;
            a.x = (row >= 0) ? Cb[row][kk * 4 + 2 * hi]     : 0.0f;
            a.y = (row >= 0) ? Cb[row][kk * 4 + 2 * hi + 1] : 0.0f;
            for (int nt = 0; nt < 4; ++nt) {
                v2f b = ldBT(&qLDS[0][0], PSTRIDE, kk * 4, nt * 16, r16, hi);
                acc[nt] = wmma4(a, b, acc[nt]);
            }
        }
        __syncthreads();                    // all reads of Cb/qLDS done
        for (int nt = 0; nt < 4; ++nt)
            stD(&Cb[0][0], PSTRIDE, m0, nt * 16, r16, hi, acc[nt]);
        __syncthreads();
    }
    for (int i = tid; i < NCHUNK * HDIM; i += blockDim.x)
        carr[i] = Cb[i >> 6][i & 63];
}

// ============================================================================
// K3: h[c, r, :] = carr[c,:] @ (A^{r+1})^T + Y[c, r, :]   (one WG per r)
// ============================================================================
__global__ __launch_bounds__(256) void lru_k3_out(
    const float* __restrict__ P, const float* __restrict__ carr,
    const float* __restrict__ Y, float* __restrict__ out) {
    const int r = blockIdx.x;
    const int tid = threadIdx.x;
    const int lane = tid & 31, w = tid >> 5;
    const int r16 = lane & 15, hi = lane >> 4;
    __shared__ float prLDS[HDIM][PSTRIDE];

    stage64(P + (size_t)(r + 1) * HDIM * HDIM, prLDS, tid);   // async issue
    const int m0 = w * 16;  // stripe over chunk index c

    v8f acc[4];
    for (int nt = 0; nt < 4; ++nt) {        // accumulator preloaded with Y[c,r,:]
        v8f a;
        for (int j = 0; j < 8; ++j) {
            const int cc = m0 + j + 8 * hi;
            a[j] = Y[((size_t)cc * LCHUNK + r) * HDIM + nt * 16 + r16];
        }
        acc[nt] = a;
    }
    wait_async0();
    __syncthreads();                        // prLDS ready
    for (int kk = 0; kk < 16; ++kk) {
        v2f a = ldA(carr, HDIM, m0, kk * 4, r16, hi);
        for (int nt = 0; nt < 4; ++nt) {
            v2f b = ldBT(&prLDS[0][0], PSTRIDE, kk * 4, nt * 16, r16, hi);
            acc[nt] = wmma4(a, b, acc[nt]);
        }
    }
    for (int nt = 0; nt < 4; ++nt)
        for (int j = 0; j < 8; ++j) {
            const int cc = m0 + j + 8 * hi;
            out[((size_t)cc * LCHUNK + r) * HDIM + nt * 16 + r16] = acc[nt][j];
        }
}

// ============================================================================
extern "C" void kernel_launch(void* const* d_in, const int* in_sizes, int n_in,
                              void* d_out, int out_size, void* d_ws, size_t ws_size,
                              hipStream_t stream) {
    const float* x    = (const float*)d_in[0];  // [T, X]
    const float* h0   = (const float*)d_in[1];  // [H]
    const float* Araw = (const float*)d_in[2];  // [H, H]
    const float* Bm   = (const float*)d_in[3];  // [H, X]
    const float* cvec = (const float*)d_in[4];  // [H]

    float* ws   = (float*)d_ws;
    float* P    = ws;                          // 65 * 4096 floats (A^0..A^64)
    float* Q    = P + 65 * HDIM * HDIM;        // 7 * 4096  (A^(64*2^k))
    float* g    = Q + 7 * HDIM * HDIM;         // NC * H
    float* carr = g + NCHUNK * HDIM;           // NC * H
    float* Y    = carr + NCHUNK * HDIM;        // T * H

    hipLaunchKernelGGL(lru_k0_powers, dim3(1), dim3(256), 0, stream,
                       Araw, h0, P, Q, g);
    hipLaunchKernelGGL(lru_k1_local, dim3(NCHUNK), dim3(256), 0, stream,
                       x, Bm, cvec, P, Y, g);
    hipLaunchKernelGGL(lru_k2_carry, dim3(1), dim3(256), 0, stream,
                       Q, g, carr);
    hipLaunchKernelGGL(lru_k3_out, dim3(LCHUNK), dim3(256), 0, stream,
                       P, carr, Y, (float*)d_out);
}